// FraudDetectionHybrid_42700564857540
// MI455X (gfx1250) — compile-verified
//
#include <hip/hip_runtime.h>
#include <cstdint>

#define LAYERS 8

typedef float v2f __attribute__((ext_vector_type(2)));
typedef float v4f __attribute__((ext_vector_type(4)));
typedef float v8f __attribute__((ext_vector_type(8)));

// gfx1250 hardware tanh (transcendental unit, co-executes with VALU)
__device__ __forceinline__ float htanh(float x) {
#if __has_builtin(__builtin_amdgcn_tanhf)
  return __builtin_amdgcn_tanhf(x);
#else
  float r;
  asm("v_tanh_f32 %0, %1" : "=v"(r) : "v"(x));
  return r;
#endif
}

// CDNA5 async global->LDS copy (ASYNCcnt-tracked, no VGPR staging)
__device__ __forceinline__ void async_load_b128_to_lds(const float* gsrc,
                                                       uint32_t lds_byte_addr) {
  asm volatile("global_load_async_to_lds_b128 %0, %1, off"
               :
               : "v"(lds_byte_addr), "v"(gsrc)
               : "memory");
}

// CDNA5 async LDS->global store: DMA the result tile out, off the wave's
// critical path (S_ENDPGM's implicit wait-idle covers completion).
__device__ __forceinline__ void async_store_b128_from_lds(float* gdst,
                                                          uint32_t lds_byte_addr) {
  asm volatile("global_store_async_from_lds_b128 %0, %1, off"
               :
               : "v"(gdst), "v"(lds_byte_addr)
               : "memory");
}

__device__ __forceinline__ void wait_asynccnt0() {
#if __has_builtin(__builtin_amdgcn_s_wait_asynccnt)
  __builtin_amdgcn_s_wait_asynccnt(0);
#else
  asm volatile("s_wait_asynccnt 0x0" ::: "memory");
#endif
}

__device__ __forceinline__ void wait_dscnt0() {
  asm volatile("s_wait_dscnt 0x0" ::: "memory");
}

__global__ __launch_bounds__(256)
void fraud_fwd_kernel(const float* __restrict__ x,
                      const float* __restrict__ W,
                      const float* __restrict__ b,
                      const float* __restrict__ scale,
                      const float* __restrict__ shift,
                      const float* __restrict__ W_out,
                      const float* __restrict__ b_out,
                      float* __restrict__ out) {
  // 8 waves * 128 rows * 2 floats = 2048 floats = 8 KB
  __shared__ float tile[2048];

  const int tid  = threadIdx.x;
  const int wave = tid >> 5;
  const int lane = tid & 31;

  const long long blockRow = (long long)blockIdx.x * 1024;
  const long long waveRow  = blockRow + (long long)wave * 128;
  const long long gFloat   = waveRow * 2;  // float index into x[B,2]

  const int      base        = wave * 256;  // this wave's slice (floats)
  const uint32_t ldsWaveByte = (uint32_t)(uintptr_t)&tile[base];

  // Stage 128 rows per wave, fully coalesced (16 B per lane per op).
  async_load_b128_to_lds(x + gFloat + 0   + lane * 4, ldsWaveByte + 0   + lane * 16);
  async_load_b128_to_lds(x + gFloat + 128 + lane * 4, ldsWaveByte + 512 + lane * 16);

  // Uniform parameter loads -> s_load_b512 into SGPRs while the async copy flies.
  float w00[LAYERS], w01[LAYERS], w10[LAYERS], w11[LAYERS];
  float b0[LAYERS], b1[LAYERS], s0[LAYERS], s1[LAYERS], t0[LAYERS], t1[LAYERS];
#pragma unroll
  for (int l = 0; l < LAYERS; ++l) {
    w00[l] = W[l * 4 + 0];  w01[l] = W[l * 4 + 1];
    w10[l] = W[l * 4 + 2];  w11[l] = W[l * 4 + 3];
    b0[l]  = b[l * 2 + 0];  b1[l]  = b[l * 2 + 1];
    s0[l]  = scale[l * 2 + 0]; s1[l] = scale[l * 2 + 1];
    t0[l]  = shift[l * 2 + 0]; t1[l] = shift[l * 2 + 1];
  }
  const float wo0 = W_out[0], wo1 = W_out[1], bo = b_out[0];

  wait_asynccnt0();  // wave-local: our lanes consume only this wave's slice

  // Each lane pulls its own 4 rows (8 floats) back out of the wave slice.
  const v4f p0 = *(const v4f*)&tile[base + lane * 8 + 0];
  const v4f p1 = *(const v4f*)&tile[base + lane * 8 + 4];

  float a[4] = {p0.x, p0.z, p1.x, p1.z};
  float c[4] = {p0.y, p0.w, p1.y, p1.w};

  // 8-layer residual tanh core: lane-per-row, FMA + v_tanh_f32.
#pragma unroll
  for (int r = 0; r < 4; ++r) {
    float av = a[r], cv = c[r];
#pragma unroll
    for (int l = 0; l < LAYERS; ++l) {
      float z0 = fmaf(cv, w01[l], fmaf(av, w00[l], b0[l]));
      float z1 = fmaf(cv, w11[l], fmaf(av, w10[l], b1[l]));
      av += fmaf(htanh(z0), s0[l], t0[l]);
      cv += fmaf(htanh(z1), s1[l], t1[l]);
    }
    a[r] = av; c[r] = cv;
  }

  // ---- WMMA epilogue: o = h @ W_out^T + b_out via V_WMMA_F32_16X16X4_F32 ----
#if __has_builtin(__builtin_amdgcn_wmma_f32_16x16x4_f32)
  // Write final h pairs back to the (consumed) input slice: tile[base+2r]=a_r, +1=c_r.
  v4f q0 = {a[0], c[0], a[1], c[1]};
  v4f q1 = {a[2], c[2], a[3], c[3]};
  *(v4f*)&tile[base + lane * 8 + 0] = q0;
  *(v4f*)&tile[base + lane * 8 + 4] = q1;

  // A-matrix (16x4, documented layout: lanes 0-15 hold K=0/1 in VGPR0/1,
  // lanes 16-31 hold K=2/3):  A[0,:] = [wo0, wo1, bo, 0], rows 1-15 = 0.
  v2f A;
  A.x = (lane == 0) ? wo0 : ((lane == 16) ? bo : 0.0f);
  A.y = (lane == 0) ? wo1 : 0.0f;
  v8f cz = {0.0f, 0.0f, 0.0f, 0.0f, 0.0f, 0.0f, 0.0f, 0.0f};

  const bool lo  = lane < 16;
  const int  col = lane & 15;

#pragma unroll
  for (int g = 0; g < 8; ++g) {
    // B-matrix (4x16): column n in lane n; low lanes carry K=0/1 (=a_n,c_n),
    // high lanes carry K=2/3 (=1,0 bias rows). Per-wave LDS ops are in-order,
    // so this read sees the q0/q1 stores above without an explicit wait.
    v2f hc = *(const v2f*)&tile[base + g * 32 + col * 2];
    v2f Bm;
    Bm.x = lo ? hc.x : 1.0f;
    Bm.y = lo ? hc.y : 0.0f;

    // D[0,n] = wo0*a_n + wo1*c_n + bo  -> lanes 0-15 of D VGPR0; rows 1-15 = 0.
    v8f d = __builtin_amdgcn_wmma_f32_16x16x4_f32(false, A, false, Bm,
                                                  (short)0, cz, false, false);
    float oval = d[0];
    asm volatile("" : "+v"(oval));  // anchor: WMMA stays outside the masked store
    if (lo) tile[base + g * 16 + col] = oval;  // results land in tile[base+0..127]
  }

  wait_dscnt0();  // result stores visible to the async DMA engine
  async_store_b128_from_lds(out + waveRow + lane * 4, ldsWaveByte + lane * 16);
#else
  float o[4];
#pragma unroll
  for (int r = 0; r < 4; ++r) o[r] = fmaf(a[r], wo0, fmaf(c[r], wo1, bo));
  *(v4f*)(out + waveRow + lane * 4) = (v4f){o[0], o[1], o[2], o[3]};
#endif
}

extern "C" void kernel_launch(void* const* d_in, const int* in_sizes, int n_in,
                              void* d_out, int out_size, void* d_ws, size_t ws_size,
                              hipStream_t stream) {
  const float* x     = (const float*)d_in[0];
  const float* W     = (const float*)d_in[1];
  const float* b     = (const float*)d_in[2];
  const float* scale = (const float*)d_in[3];
  const float* shift = (const float*)d_in[4];
  const float* W_out = (const float*)d_in[5];
  const float* b_out = (const float*)d_in[6];
  float* out = (float*)d_out;

  const long long rows   = (long long)in_sizes[0] / 2;  // 16,777,216
  const int       blocks = (int)(rows / 1024);          // 16,384 (exact)

  fraud_fwd_kernel<<<blocks, 256, 0, stream>>>(x, W, b, scale, shift,
                                               W_out, b_out, out);
}